// BaselineModel_67989332295843
// MI455X (gfx1250) — compile-verified
//
#include <hip/hip_runtime.h>

#define D 256

typedef float v2f __attribute__((ext_vector_type(2)));
typedef float v8f __attribute__((ext_vector_type(8)));

// ---------------- CSR build ----------------

__global__ __launch_bounds__(256) void zero_i32(int* __restrict__ p, int n) {
  int i = blockIdx.x * 256 + threadIdx.x;
  if (i < n) p[i] = 0;
}

__global__ __launch_bounds__(256) void hist_kernel(const int* __restrict__ dst,
                                                   int* __restrict__ deg, int E) {
  int e = blockIdx.x * 256 + threadIdx.x;
  if (e < E) atomicAdd(&deg[dst[e]], 1);
}

// Single-block Hillis-Steele scan over N (~50k) -> row offsets, fill cursor, 1/deg.
__global__ __launch_bounds__(1024) void scan_kernel(const int* __restrict__ deg,
                                                    int* __restrict__ row_off,
                                                    int* __restrict__ cursor,
                                                    float* __restrict__ inv_deg, int n) {
  __shared__ int tmp[1024];
  __shared__ int carry;
  const int tid = threadIdx.x;
  if (tid == 0) carry = 0;
  __syncthreads();
  for (int base = 0; base < n; base += 1024) {
    int i = base + tid;
    int v = (i < n) ? deg[i] : 0;
    tmp[tid] = v;
    __syncthreads();
    for (int s = 1; s < 1024; s <<= 1) {
      int t = (tid >= s) ? tmp[tid - s] : 0;
      __syncthreads();
      tmp[tid] += t;
      __syncthreads();
    }
    int incl = tmp[tid] + carry;  // inclusive prefix including previous chunks
    if (i < n) {
      int excl = incl - v;
      row_off[i] = excl;
      cursor[i]  = excl;
      inv_deg[i] = 1.0f / (float)(v > 1 ? v : 1);
    }
    __syncthreads();
    if (tid == 1023) carry = incl;
    __syncthreads();
  }
  if (tid == 0) row_off[n] = carry;
}

__global__ __launch_bounds__(256) void fill_kernel(const int* __restrict__ src,
                                                   const int* __restrict__ dst,
                                                   int* __restrict__ cursor,
                                                   int* __restrict__ col, int E) {
  int e = blockIdx.x * 256 + threadIdx.x;
  if (e < E) {
    int p = atomicAdd(&cursor[dst[e]], 1);
    col[p] = src[e];
  }
}

// ---------------- Mean aggregation (gather form, no float atomics) ----------------
// One block per node; thread = feature lane. All h-row reads are 1KB coalesced
// and hit L2 (feature matrix = 51MB << 192MB L2).

__global__ __launch_bounds__(256) void agg_kernel(const float* __restrict__ h,
                                                  const int* __restrict__ row_off,
                                                  const int* __restrict__ col,
                                                  const float* __restrict__ inv_deg,
                                                  float* __restrict__ agg) {
  const int node = blockIdx.x;
  const int f = threadIdx.x;
  const int s = row_off[node];
  const int e = row_off[node + 1];
  float acc = 0.f;
  for (int j = s; j < e; ++j) {
    acc += h[(size_t)col[j] * D + f];
  }
  agg[(size_t)node * D + f] = acc * inv_deg[node];
}

// ---------------- Fused dual-GEMM: out = A1@W1 + A2@W2 + bias (+ReLU) ----------------
// fp32 WMMA 16x16x4. Block: 8 waves, 128(M) x 64(N) tile. Both 256x64 weight
// tiles staged in 128KB LDS, layout [k/4][col][4] so the per-lane (K,K+1) pair
// is one 8-byte ds_load. A-matrix lane layout per ISA: lanes 0-15 hold K+0/K+1,
// lanes 16-31 hold K+2/K+3, M = lane&15. C VGPR r holds rows r (lanes<16) and
// r+8 (lanes>=16), N = lane&15 within each 16-col tile.

__global__ __launch_bounds__(256) void sage_gemm(const float* __restrict__ A1,
                                                 const float* __restrict__ A2,
                                                 const float* __restrict__ W1,
                                                 const float* __restrict__ W2,
                                                 const float* __restrict__ bias,
                                                 float* __restrict__ out,
                                                 int nrows, int relu) {
  __shared__ float sW1[D * 64];
  __shared__ float sW2[D * 64];
  const int tid = threadIdx.x;
  const int colBase = blockIdx.y * 64;
  const int rowBase = blockIdx.x * 128;

  // Stage weight tiles (coalesced global reads).
  for (int i = tid; i < D * 64; i += 256) {
    int kk = i >> 6, cc = i & 63;
    int si = ((kk >> 2) * 64 + cc) * 4 + (kk & 3);
    sW1[si] = W1[kk * D + colBase + cc];
    sW2[si] = W2[kk * D + colBase + cc];
  }
  __syncthreads();

  const int wave  = tid >> 5;
  const int lane  = tid & 31;
  const int laneM = lane & 15;
  const int kOff  = (lane >> 4) << 1;   // 0 or 2

  int row0 = rowBase + wave * 16;
  if (row0 >= nrows) return;            // after the only barrier; safe
  if (row0 + 16 > nrows) row0 = nrows - 16;

  const float* a1p = A1 + (size_t)(row0 + laneM) * D + kOff;
  const float* a2p = A2 + (size_t)(row0 + laneM) * D + kOff;

  v8f c0 = {}, c1 = {}, c2 = {}, c3 = {};

#pragma unroll 2
  for (int k = 0; k < D; k += 4) {
    v2f a1 = *(const v2f*)(a1p + k);
    v2f a2 = *(const v2f*)(a2p + k);
    const float* s1 = sW1 + (k >> 2) * 256 + kOff;
    const float* s2 = sW2 + (k >> 2) * 256 + kOff;
    v2f b;

    b  = *(const v2f*)(s1 + (0 * 16 + laneM) * 4);
    c0 = __builtin_amdgcn_wmma_f32_16x16x4_f32(false, a1, false, b, (short)0, c0, false, false);
    b  = *(const v2f*)(s2 + (0 * 16 + laneM) * 4);
    c0 = __builtin_amdgcn_wmma_f32_16x16x4_f32(false, a2, false, b, (short)0, c0, false, false);

    b  = *(const v2f*)(s1 + (1 * 16 + laneM) * 4);
    c1 = __builtin_amdgcn_wmma_f32_16x16x4_f32(false, a1, false, b, (short)0, c1, false, false);
    b  = *(const v2f*)(s2 + (1 * 16 + laneM) * 4);
    c1 = __builtin_amdgcn_wmma_f32_16x16x4_f32(false, a2, false, b, (short)0, c1, false, false);

    b  = *(const v2f*)(s1 + (2 * 16 + laneM) * 4);
    c2 = __builtin_amdgcn_wmma_f32_16x16x4_f32(false, a1, false, b, (short)0, c2, false, false);
    b  = *(const v2f*)(s2 + (2 * 16 + laneM) * 4);
    c2 = __builtin_amdgcn_wmma_f32_16x16x4_f32(false, a2, false, b, (short)0, c2, false, false);

    b  = *(const v2f*)(s1 + (3 * 16 + laneM) * 4);
    c3 = __builtin_amdgcn_wmma_f32_16x16x4_f32(false, a1, false, b, (short)0, c3, false, false);
    b  = *(const v2f*)(s2 + (3 * 16 + laneM) * 4);
    c3 = __builtin_amdgcn_wmma_f32_16x16x4_f32(false, a2, false, b, (short)0, c3, false, false);
  }

  const int mAdd = (lane >> 4) * 8;
#pragma unroll
  for (int n = 0; n < 4; ++n) {
    v8f c = (n == 0) ? c0 : (n == 1) ? c1 : (n == 2) ? c2 : c3;
    int colG = colBase + n * 16 + laneM;
    float bv = bias[colG];
#pragma unroll
    for (int r = 0; r < 8; ++r) {
      float v = c[r] + bv;
      if (relu) v = fmaxf(v, 0.f);
      out[(size_t)(row0 + mAdd + r) * D + colG] = v;
    }
  }
}

// ---------------- Host orchestration ----------------

extern "C" void kernel_launch(void* const* d_in, const int* in_sizes, int n_in,
                              void* d_out, int out_size, void* d_ws, size_t ws_size,
                              hipStream_t stream) {
  const float* x  = (const float*)d_in[0];
  const int*   ei = (const int*)d_in[1];
  const float* Wl = (const float*)d_in[2];
  const float* Wr = (const float*)d_in[3];
  const float* bb = (const float*)d_in[4];

  const int N = in_sizes[0] / D;
  const int E = in_sizes[1] / 2;
  const int L = in_sizes[4] / D;

  const int* src = ei;       // edge_index[0]
  const int* dst = ei + E;   // edge_index[1]

  char* ws = (char*)d_ws;
  size_t off = 0;
  auto alloc = [&](size_t bytes) -> char* {
    char* p = ws + off;
    off = (off + bytes + 255) & ~(size_t)255;
    return p;
  };
  int*   deg     = (int*)  alloc((size_t)N * 4);
  int*   row_off = (int*)  alloc((size_t)(N + 1) * 4);
  int*   cursor  = (int*)  alloc((size_t)N * 4);
  float* inv_deg = (float*)alloc((size_t)N * 4);
  int*   col     = (int*)  alloc((size_t)E * 4);
  float* agg     = (float*)alloc((size_t)N * D * 4);
  float* hA      = (float*)alloc((size_t)N * D * 4);
  float* hB      = (float*)alloc((size_t)N * D * 4);
  (void)ws_size; (void)n_in; (void)out_size;

  // CSR build (deterministic per call; rebuilt every call, no cached state)
  zero_i32<<<(N + 255) / 256, 256, 0, stream>>>(deg, N);
  hist_kernel<<<(E + 255) / 256, 256, 0, stream>>>(dst, deg, E);
  scan_kernel<<<1, 1024, 0, stream>>>(deg, row_off, cursor, inv_deg, N);
  fill_kernel<<<(E + 255) / 256, 256, 0, stream>>>(src, dst, cursor, col, E);

  const float* h_in = x;
  for (int l = 0; l < L; ++l) {
    float* h_out = (l == L - 1) ? (float*)d_out : ((l & 1) ? hB : hA);
    agg_kernel<<<N, 256, 0, stream>>>(h_in, row_off, col, inv_deg, agg);
    dim3 grid((N + 127) / 128, D / 64);
    sage_gemm<<<grid, 256, 0, stream>>>(agg, h_in,
                                        Wl + (size_t)l * D * D,
                                        Wr + (size_t)l * D * D,
                                        bb + (size_t)l * D,
                                        h_out, N, (l < L - 1) ? 1 : 0);
    h_in = h_out;
  }
}